// GNNEncoder_13142599925846
// MI455X (gfx1250) — compile-verified
//
#include <hip/hip_runtime.h>

// ---------------------------------------------------------------------------
// GraphSAGE 3-layer forward for MI455X (gfx1250, wave32, WMMA)
//   layer: out = relu?( (segsum(x[src]->dst)*inv_deg) @ Wl^T + bl + x @ Wr^T )
// fp32 end-to-end: dense GEMMs use V_WMMA_F32_16X16X4_F32 (exact fp32 MAC).
// ---------------------------------------------------------------------------

typedef __attribute__((ext_vector_type(2))) float v2f;
typedef __attribute__((ext_vector_type(8))) float v8f;

#define N_NODES 100000
#define N_EDGES 1600000
#define D_IN    128
#define D_HID   256
#define D_OUT   128

// ---------------------------------------------------------------- zero fill
__global__ void zero_f32_kernel(float* __restrict__ p, long n) {
    long i = (long)blockIdx.x * blockDim.x + threadIdx.x;
    long stride = (long)gridDim.x * blockDim.x;
    for (; i < n; i += stride) p[i] = 0.0f;
}

// ---------------------------------------------------------------- degree
__global__ void degree_kernel(const int* __restrict__ dst, float* __restrict__ deg, int nE) {
    int i = blockIdx.x * blockDim.x + threadIdx.x;
    int stride = gridDim.x * blockDim.x;
    for (; i < nE; i += stride) atomicAdd(&deg[dst[i]], 1.0f);
}

__global__ void invdeg_kernel(float* __restrict__ deg, int n) {
    int i = blockIdx.x * blockDim.x + threadIdx.x;
    if (i < n) {
        float d = deg[i];
        deg[i] = (d > 0.0f) ? (1.0f / d) : 0.0f;   // matches where(deg>0, 1/max(deg,1), 0)
    }
}

// ---------------------------------------------------------------- scatter add
// One wave per edge per iteration: lanes gather a full feature row of x[src]
// as float4 (coalesced 512B per 128-float chunk) and atomic-add into agg[dst] (L2).
template<int F>
__global__ __launch_bounds__(256)
void scatter_add_kernel(const float* __restrict__ x, const int* __restrict__ src,
                        const int* __restrict__ dst, float* __restrict__ agg, int nE) {
    int gwave = (blockIdx.x * blockDim.x + threadIdx.x) >> 5;
    int lane  = threadIdx.x & 31;
    int nW    = (gridDim.x * blockDim.x) >> 5;
    for (int e = gwave; e < nE; e += nW) {
        int s = src[e];
        int d = dst[e];
        const float4* xs = (const float4*)(x + (size_t)s * F);
        float*        ad = agg + (size_t)d * F;
#pragma unroll
        for (int c = 0; c < F / 128; ++c) {
            int q = lane + 32 * c;          // float4 index within the row
            float4 v = xs[q];
            int b = q * 4;
            atomicAdd(ad + b + 0, v.x);
            atomicAdd(ad + b + 1, v.y);
            atomicAdd(ad + b + 2, v.z);
            atomicAdd(ad + b + 3, v.w);
        }
    }
}

// ---------------------------------------------------------------- SAGE GEMM
// Block = 256 threads = 8 waves. Block computes a 16-row (node) slab.
// Wave w handles output column tile j0 = (blockIdx.y*8 + w)*16.
// A tiles (mean = agg*inv_deg, and self = x) staged in LDS once per block.
// K-loop step 4: two v_wmma_f32_16x16x4_f32 per step (mean*Wl^T, x*Wr^T).
//
// Fragment layouts per CDNA5 ISA 7.12.2 (wave32):
//   A 16x4 f32 : lane L -> row m=L%16 ; VGPR0/1 = K = 2*(L/16) + {0,1}
//   B 4x16 f32 : lane L -> col n=L%16 ; VGPR0/1 = K = 2*(L/16) + {0,1}
//   C/D 16x16  : lane L -> col n=L%16 ; VGPR r  = row m = r + 8*(L/16)
template<int K, bool RELU>
__global__ __launch_bounds__(256)
void sage_gemm_kernel(const float* __restrict__ agg,      // [N,K] unscaled sums
                      const float* __restrict__ inv_deg,  // [N]
                      const float* __restrict__ xself,    // [N,K]
                      const float* __restrict__ Wl,       // [Fout,K] row-major
                      const float* __restrict__ bl,       // [Fout]
                      const float* __restrict__ Wr,       // [Fout,K] row-major
                      float* __restrict__ out,            // [N,Fout]
                      int Fout) {
    constexpr int PAD = 4;
    constexpr int LDK = K + PAD;
    __shared__ float smMean[16 * LDK];
    __shared__ float smSelf[16 * LDK];

    const int  tid  = threadIdx.x;
    const long row0 = (long)blockIdx.x * 16;

    // Cooperative staging: 16 x K floats of each A matrix, scale mean on load.
    for (int idx = tid; idx < 16 * K; idx += 256) {
        int m = idx / K;            // K is a power of two -> shifts
        int k = idx % K;
        long node = row0 + m;
        float s = inv_deg[node];
        smMean[m * LDK + k] = agg[node * K + k] * s;
        smSelf[m * LDK + k] = xself[node * K + k];
    }
    __syncthreads();

    const int wave = tid >> 5;
    const int lane = tid & 31;
    const int n    = lane & 15;     // output column within tile / A row
    const int half = lane >> 4;     // 0: K pair {0,1}; 1: K pair {2,3}
    const int koff = half * 2;

    const int j = (blockIdx.y * 8 + wave) * 16 + n;   // global output column
    const float* wlRow  = Wl + (long)j * K + koff;    // contiguous K in memory
    const float* wrRow  = Wr + (long)j * K + koff;
    const float* aMeanP = smMean + n * LDK + koff;    // row m == n for A frags
    const float* aSelfP = smSelf + n * LDK + koff;

    v8f acc = {};
#pragma unroll 4
    for (int k0 = 0; k0 < K; k0 += 4) {
        v2f am = *(const v2f*)(aMeanP + k0);          // ds_load_b64
        v2f as = *(const v2f*)(aSelfP + k0);
        v2f bL = *(const v2f*)(wlRow + k0);           // global_load_b64
        v2f bR = *(const v2f*)(wrRow + k0);
        // D = A*B + C, fp32 exact
        acc = __builtin_amdgcn_wmma_f32_16x16x4_f32(false, am, false, bL,
                                                    (short)0, acc, false, false);
        acc = __builtin_amdgcn_wmma_f32_16x16x4_f32(false, as, false, bR,
                                                    (short)0, acc, false, false);
    }

    const float bias = bl[j];
#pragma unroll
    for (int r = 0; r < 8; ++r) {
        long node = row0 + r + 8 * half;
        float v = acc[r] + bias;
        if (RELU) v = fmaxf(v, 0.0f);
        out[node * Fout + j] = v;
    }
}

// ---------------------------------------------------------------- launch
extern "C" void kernel_launch(void* const* d_in, const int* in_sizes, int n_in,
                              void* d_out, int out_size, void* d_ws, size_t ws_size,
                              hipStream_t stream) {
    const float* x    = (const float*)d_in[0];
    const int*   edge = (const int*)d_in[1];
    const int*   src  = edge;
    const int*   dst  = edge + N_EDGES;
    const float* Wl1 = (const float*)d_in[2];
    const float* bl1 = (const float*)d_in[3];
    const float* Wr1 = (const float*)d_in[4];
    const float* Wl2 = (const float*)d_in[5];
    const float* bl2 = (const float*)d_in[6];
    const float* Wr2 = (const float*)d_in[7];
    const float* Wl3 = (const float*)d_in[8];
    const float* bl3 = (const float*)d_in[9];
    const float* Wr3 = (const float*)d_in[10];
    float* out = (float*)d_out;

    // Workspace layout (floats): inv_deg[N] | agg[N*256] | h1[N*256] | h2[N*256]
    float* invdeg = (float*)d_ws;
    float* agg    = invdeg + N_NODES;
    float* h1     = agg + (size_t)N_NODES * D_HID;
    float* h2     = h1  + (size_t)N_NODES * D_HID;

    const dim3 blk(256);

    // --- degree -> inv_deg ---
    zero_f32_kernel<<<2048, blk, 0, stream>>>(invdeg, (long)N_NODES);
    degree_kernel<<<2048, blk, 0, stream>>>(dst, invdeg, N_EDGES);
    invdeg_kernel<<<(N_NODES + 255) / 256, blk, 0, stream>>>(invdeg, N_NODES);

    // --- layer 1: D_IN=128 -> D_HID=256, relu ---
    zero_f32_kernel<<<4096, blk, 0, stream>>>(agg, (long)N_NODES * D_IN);
    scatter_add_kernel<D_IN><<<4096, blk, 0, stream>>>(x, src, dst, agg, N_EDGES);
    sage_gemm_kernel<D_IN, true><<<dim3(N_NODES / 16, D_HID / 128), blk, 0, stream>>>(
        agg, invdeg, x, Wl1, bl1, Wr1, h1, D_HID);

    // --- layer 2: 256 -> 256, relu ---
    zero_f32_kernel<<<4096, blk, 0, stream>>>(agg, (long)N_NODES * D_HID);
    scatter_add_kernel<D_HID><<<4096, blk, 0, stream>>>(h1, src, dst, agg, N_EDGES);
    sage_gemm_kernel<D_HID, true><<<dim3(N_NODES / 16, D_HID / 128), blk, 0, stream>>>(
        agg, invdeg, h1, Wl2, bl2, Wr2, h2, D_HID);

    // --- layer 3: 256 -> D_OUT=128, no relu ---
    zero_f32_kernel<<<4096, blk, 0, stream>>>(agg, (long)N_NODES * D_HID);
    scatter_add_kernel<D_HID><<<4096, blk, 0, stream>>>(h2, src, dst, agg, N_EDGES);
    sage_gemm_kernel<D_HID, false><<<dim3(N_NODES / 16, D_OUT / 128), blk, 0, stream>>>(
        agg, invdeg, h2, Wl3, bl3, Wr3, out, D_OUT);
}